// OnlineBootstrappingLoss_25348896981481
// MI455X (gfx1250) — compile-verified
//
#include <hip/hip_runtime.h>
#include <stdint.h>

#define B_  16
#define C_  8
#define HW_ 262144           // 512*512
#define N_  (B_ * HW_)
#define KEEP_ 512
#define SEL_THREADS 1024
#define TILE_ELEMS (SEL_THREADS * 4)   // b128 per lane: 4 floats
#define TIE_CAP 256

typedef __attribute__((ext_vector_type(2))) float v2f;
typedef __attribute__((ext_vector_type(8))) float v8f;

// ---------------------------------------------------------------------------
// Phase 1: per-pixel L1-distance-to-onehot and NLL (logsumexp - x_t).
// 4 pixels per thread -> all global traffic is b128 (144 MiB in, 32 MiB out).
// ---------------------------------------------------------------------------
__global__ __launch_bounds__(256) void dist_nll_kernel(
    const float* __restrict__ x, const int* __restrict__ tgt,
    float* __restrict__ dist, float* __restrict__ nll) {
  int q = blockIdx.x * blockDim.x + threadIdx.x;   // quad-pixel index
  if (q >= N_ / 4) return;
  int g0 = q * 4;
  int b  = g0 >> 18;          // / HW_
  int p0 = g0 & (HW_ - 1);
  const float* xb = x + (size_t)b * C_ * HW_ + p0;

  float vv[C_][4];
#pragma unroll
  for (int c = 0; c < C_; ++c) {
    float4 f = *(const float4*)(xb + (size_t)c * HW_);   // global_load_b128
    vv[c][0] = f.x; vv[c][1] = f.y; vv[c][2] = f.z; vv[c][3] = f.w;
  }
  int4 t4 = *(const int4*)(tgt + g0);
  int tt[4] = {t4.x, t4.y, t4.z, t4.w};

  float od[4], on[4];
#pragma unroll
  for (int j = 0; j < 4; ++j) {
    float sabs = 0.f, m = vv[0][j], xt = 0.f;
#pragma unroll
    for (int c = 0; c < C_; ++c) {
      float val = vv[c][j];
      sabs += fabsf(val);
      m = fmaxf(m, val);
      if (c == tt[j]) xt = val;
    }
    float se = 0.f;
#pragma unroll
    for (int c = 0; c < C_; ++c) se += expf(vv[c][j] - m);
    od[j] = sabs - fabsf(xt) + fabsf(xt - 1.f);
    on[j] = m + logf(se) - xt;
  }
  *(float4*)(dist + g0) = make_float4(od[0], od[1], od[2], od[3]);
  *(float4*)(nll  + g0) = make_float4(on[0], on[1], on[2], on[3]);
}

// ---------------------------------------------------------------------------
// Async global->LDS (CDNA5): issue without waiting; split waits allow
// double-buffered pipelining (issue tile i+1, wait cnt<=1 => tile i landed).
// ---------------------------------------------------------------------------
__device__ __forceinline__ void async_issue_b128(uint32_t lds_byte_addr,
                                                 unsigned long long gaddr) {
  asm volatile("global_load_async_to_lds_b128 %0, %1, off"
               :: "v"(lds_byte_addr), "v"(gaddr) : "memory");
}
__device__ __forceinline__ void async_wait0() {
  asm volatile("s_wait_asynccnt 0x0" ::: "memory");
}
__device__ __forceinline__ void async_wait1() {
  asm volatile("s_wait_asynccnt 0x1" ::: "memory");
}

// ---------------------------------------------------------------------------
// Phase 2: exact per-sample top-K via 3-pass radix select on float bits
// (dist >= 0 so u32 bit pattern is order-preserving), then CE accumulation
// with lowest-index tie-break (matches jax.lax.top_k).
// ---------------------------------------------------------------------------
__global__ __launch_bounds__(SEL_THREADS) void select_kernel(
    const float* __restrict__ dist, const float* __restrict__ nll,
    float* __restrict__ samp_sum) {
  __shared__ unsigned int hist[2048];
  __shared__ float tile[2][TILE_ELEMS];    // double-buffered async tiles (32 KB)
  __shared__ int tie_idx[TIE_CAP];
  __shared__ unsigned int sh_tiecnt;
  __shared__ unsigned int sh_b, sh_r;
  __shared__ float sh_tot;
  __shared__ float wpart[32];

  const int s = blockIdx.x;
  const int tid = threadIdx.x;
  const float* ds = dist + (size_t)s * HW_;
  const float* ns = nll  + (size_t)s * HW_;

  const uint32_t lds_t[2] = {(uint32_t)(uintptr_t)&tile[0][tid * 4],
                             (uint32_t)(uintptr_t)&tile[1][tid * 4]};

  if (tid == 0) sh_r = KEEP_;

  // radix-select: bits [31:21], [20:10], [9:0]
  const int shifts[3] = {21, 10, 0};
  const int nbins[3]  = {2048, 2048, 1024};
  const int fbits[3]  = {11, 11, 10};

  uint32_t pref = 0;
  int pshift = 32;  // 32 => no prefix constraint yet

  for (int pass = 0; pass < 3; ++pass) {
    for (int i = tid; i < 2048; i += SEL_THREADS) hist[i] = 0u;
    __syncthreads();

    const int shift = shifts[pass];
    const uint32_t mask = (uint32_t)nbins[pass] - 1u;

    async_issue_b128(lds_t[0], (unsigned long long)(uintptr_t)(ds + tid * 4));
    int cur = 0;
    for (int base = 0; base < HW_; base += TILE_ELEMS) {
      int nxt = base + TILE_ELEMS;
      if (nxt < HW_) {
        async_issue_b128(lds_t[cur ^ 1],
                         (unsigned long long)(uintptr_t)(ds + nxt + tid * 4));
        async_wait1();            // tile `cur` has landed
      } else {
        async_wait0();
      }
#pragma unroll
      for (int k = 0; k < 4; ++k) {
        uint32_t key = __float_as_uint(tile[cur][tid * 4 + k]);
        if (pshift >= 32 || (key >> pshift) == pref)
          atomicAdd(&hist[(key >> shift) & mask], 1u);
      }
      cur ^= 1;
    }
    __syncthreads();

    if (tid == 0) {
      uint32_t r = sh_r, cum = 0;
      int bsel = 0;
      for (int bin = nbins[pass] - 1; bin >= 0; --bin) {
        uint32_t c = hist[bin];
        if (cum + c >= r) { bsel = bin; break; }
        cum += c;
      }
      sh_b = (uint32_t)bsel;
      sh_r = r - cum;           // rank within selected bin
    }
    __syncthreads();
    pref = (pref << fbits[pass]) | sh_b;
    pshift = shift;
    __syncthreads();
  }

  const uint32_t T = pref;      // exact K-th largest key
  if (tid == 0) sh_tiecnt = 0u;
  __syncthreads();

  // Final pass: accumulate nll for key > T, record tie indices (key == T).
  float acc = 0.f;
  {
    async_issue_b128(lds_t[0], (unsigned long long)(uintptr_t)(ds + tid * 4));
    int cur = 0;
    for (int base = 0; base < HW_; base += TILE_ELEMS) {
      int nxt = base + TILE_ELEMS;
      if (nxt < HW_) {
        async_issue_b128(lds_t[cur ^ 1],
                         (unsigned long long)(uintptr_t)(ds + nxt + tid * 4));
        async_wait1();
      } else {
        async_wait0();
      }
      int p0 = base + tid * 4;
#pragma unroll
      for (int k = 0; k < 4; ++k) {
        uint32_t key = __float_as_uint(tile[cur][tid * 4 + k]);
        if (key > T) {
          acc += ns[p0 + k];
        } else if (key == T) {
          unsigned int slot = atomicAdd(&sh_tiecnt, 1u);
          if (slot < TIE_CAP) tie_idx[slot] = p0 + k;
        }
      }
      cur ^= 1;
    }
  }
  __syncthreads();

  // Block reduce: wave-level shuffles, then WMMA over the 32 wave partials.
  for (int d = 16; d > 0; d >>= 1) acc += __shfl_down(acc, d, 32);
  if ((tid & 31) == 0) wpart[tid >> 5] = acc;
  __syncthreads();

  if (tid < 32) {               // wave 0, EXEC all-ones (WMMA requirement)
    float xv = wpart[tid];
#if __has_builtin(__builtin_amdgcn_wmma_f32_16x16x4_f32)
    // A(16x4): lane m -> row m K=0 (VGPR0), lane m+16 -> row m K=2; K=1,3 zero.
    // B(4x16) = ones  =>  D[m][n] = x[m] + x[m+16].
    v2f a; a.x = xv;  a.y = 0.f;
    v2f bb; bb.x = 1.f; bb.y = 1.f;
    v8f c = {};
    c = __builtin_amdgcn_wmma_f32_16x16x4_f32(false, a, false, bb,
                                              (short)0, c, false, false);
    float srow = c[0] + c[1] + c[2] + c[3] + c[4] + c[5] + c[6] + c[7];
    float tot = srow + __shfl(srow, 16, 32);   // lower + upper half-sums
#else
    float tot = xv;
    for (int d = 16; d > 0; d >>= 1) tot += __shfl_down(tot, d, 32);
#endif
    if (tid == 0) sh_tot = tot;
  }
  __syncthreads();

  if (tid == 0) {
    float tot = sh_tot;
    unsigned int tc = sh_tiecnt < TIE_CAP ? sh_tiecnt : TIE_CAP;
    unsigned int need = sh_r;   // ties to take, lowest index first
    int last = -1;
    for (unsigned int it = 0; it < need; ++it) {
      int best = 0x7fffffff;
      for (unsigned int i = 0; i < tc; ++i) {
        int v = tie_idx[i];
        if (v > last && v < best) best = v;
      }
      if (best == 0x7fffffff) break;
      tot += ns[best];
      last = best;
    }
    samp_sum[s] = tot;
  }
}

// ---------------------------------------------------------------------------
// Phase 3: mean over B*KEEP.
// ---------------------------------------------------------------------------
__global__ void finalize_kernel(const float* __restrict__ samp,
                                float* __restrict__ out) {
  if (threadIdx.x == 0 && blockIdx.x == 0) {
    float t = 0.f;
    for (int i = 0; i < B_; ++i) t += samp[i];
    out[0] = t / (float)(B_ * KEEP_);
  }
}

extern "C" void kernel_launch(void* const* d_in, const int* in_sizes, int n_in,
                              void* d_out, int out_size, void* d_ws, size_t ws_size,
                              hipStream_t stream) {
  const float* x  = (const float*)d_in[0];
  const int* tgt  = (const int*)d_in[1];
  float* dist = (float*)d_ws;              // N_ floats (16 MiB)
  float* nllv = dist + N_;                 // N_ floats (16 MiB)
  float* samp = nllv + N_;                 // B_ floats

  dist_nll_kernel<<<(N_ / 4) / 256, 256, 0, stream>>>(x, tgt, dist, nllv);
  select_kernel<<<B_, SEL_THREADS, 0, stream>>>(dist, nllv, samp);
  finalize_kernel<<<1, 32, 0, stream>>>(samp, (float*)d_out);
}